// MMD_loss_10239202034248
// MI455X (gfx1250) — compile-verified
//
#include <hip/hip_runtime.h>
#include <hip/hip_bf16.h>

#define NSRC 4096
#define NTGT 4096
#define NTOT 8192
#define D    128

#define BM 128
#define BN 128
#define KC 32
#define LSTR 36   // KC + 4 floats pad: kills LDS bank conflicts (gcd(36,64)=4, 16-cycle)

typedef __attribute__((ext_vector_type(2))) float v2f;
typedef __attribute__((ext_vector_type(8))) float v8f;

// workspace layout (floats)
#define WS_SQ   0      // 8192: per-row squared norms of total
#define WS_COEF 8192   // 5: -log2(e)/(bw*2^i)
#define WS_SUMS 8200   // 4: region sums XX, XY, YX, YY

// ---------------- pass 1: per-row squared norms (wave per row) ----------------
__global__ __launch_bounds__(256) void sq_kernel(const float* __restrict__ src,
                                                 const float* __restrict__ tgt,
                                                 float* __restrict__ ws) {
    const int wave = threadIdx.x >> 5;
    const int lane = threadIdx.x & 31;
    const int row  = blockIdx.x * 8 + wave;
    if (row >= NTOT) return;
    const float* rp = (row < NSRC) ? (src + (size_t)row * D)
                                   : (tgt + (size_t)(row - NSRC) * D);
    float4 v = ((const float4*)rp)[lane];             // 32 lanes x float4 = 128 floats
    float s = v.x * v.x + v.y * v.y + v.z * v.z + v.w * v.w;
    #pragma unroll
    for (int off = 16; off > 0; off >>= 1) s += __shfl_xor(s, off, 32);
    if (lane == 0) ws[WS_SQ + row] = s;
}

// ---- pass 2: bandwidth via closed form  sum(l2) = 2n*sum(sq) - 2*||colsum||^2 ----
__global__ __launch_bounds__(256) void bw_kernel(const float* __restrict__ src,
                                                 const float* __restrict__ tgt,
                                                 float* __restrict__ ws) {
    __shared__ float red[256];
    __shared__ float colsq[128];
    const int t = threadIdx.x;

    float s = 0.f;
    for (int i = t; i < NTOT; i += 256) s += ws[WS_SQ + i];
    red[t] = s;
    __syncthreads();
    for (int st = 128; st > 0; st >>= 1) {
        if (t < st) red[t] += red[t + st];
        __syncthreads();
    }
    const float ssq = red[0];

    if (t < 128) {  // column sums over all 8192 rows (coalesced across the 128 lanes)
        float cs = 0.f;
        for (int r = 0; r < NSRC; ++r) cs += src[(size_t)r * D + t];
        for (int r = 0; r < NTGT; ++r) cs += tgt[(size_t)r * D + t];
        colsq[t] = cs * cs;
    }
    __syncthreads();

    if (t == 0) {
        double snorm = 0.0;
        for (int d = 0; d < 128; ++d) snorm += (double)colsq[d];
        const double n = (double)NTOT;
        double sum_l2 = 2.0 * n * (double)ssq - 2.0 * snorm;
        double bw = sum_l2 / (n * n - n);
        bw /= 4.0;  // KERNEL_MUL^(KERNEL_NUM//2) = 2^2
        const double LOG2E = 1.4426950408889634;
        for (int b = 0; b < 5; ++b)
            ws[WS_COEF + b] = (float)(-LOG2E / (bw * (double)(1 << b)));
        ws[WS_SUMS + 0] = 0.f; ws[WS_SUMS + 1] = 0.f;
        ws[WS_SUMS + 2] = 0.f; ws[WS_SUMS + 3] = 0.f;
    }
}

// ------- pass 3: fused WMMA fp32 Gram tile + multi-bandwidth Gaussian epilogue -------
// Symmetric matrix: only tiles with jb >= ib are computed; off-diagonal tile sums are
// credited to both the (i,j) region and the mirrored (j,i) region -> ~2x less work.
__global__ __launch_bounds__(256) void mmd_kernel(const float* __restrict__ src,
                                                  const float* __restrict__ tgt,
                                                  float* __restrict__ ws) {
    if (blockIdx.x < blockIdx.y) return;   // lower triangle: whole block exits, no barriers

    __shared__ __align__(16) float lA[BM * LSTR];
    __shared__ __align__(16) float lB[BN * LSTR];
    __shared__ float sqA[BM];
    __shared__ float sqB[BN];
    __shared__ float red[256];

    const int t     = threadIdx.x;
    const int lane  = t & 31;
    const int wave  = t >> 5;
    const int waveM = wave >> 1;   // 0..3 : 32-row strip
    const int waveN = wave & 1;    // 0..1 : 64-col strip
    const int iBase = blockIdx.y * BM;
    const int jBase = blockIdx.x * BN;

    // 4096 % 128 == 0 -> block lies entirely on one side of the src/tgt boundary
    const float* aBase = (iBase < NSRC) ? (src + (size_t)iBase * D)
                                        : (tgt + (size_t)(iBase - NSRC) * D);
    const float* bBase = (jBase < NSRC) ? (src + (size_t)jBase * D)
                                        : (tgt + (size_t)(jBase - NSRC) * D);

    const float c0 = ws[WS_COEF + 0], c1 = ws[WS_COEF + 1], c2 = ws[WS_COEF + 2],
                c3 = ws[WS_COEF + 3], c4 = ws[WS_COEF + 4];

    if (t < 128) sqA[t] = ws[WS_SQ + iBase + t];
    else         sqB[t - 128] = ws[WS_SQ + jBase + (t - 128)];

    const v8f vzero = {0.f, 0.f, 0.f, 0.f, 0.f, 0.f, 0.f, 0.f};
    v8f acc[2][4];
    #pragma unroll
    for (int mt = 0; mt < 2; ++mt)
        #pragma unroll
        for (int nt = 0; nt < 4; ++nt) acc[mt][nt] = vzero;

    const int kl = (lane >> 4) << 1;   // A/B frag K-interleave: lanes 16-31 hold K+2
    for (int kc = 0; kc < D; kc += KC) {
        __syncthreads();
        #pragma unroll
        for (int i = 0; i < 4; ++i) {   // 1024 float4 slots per tile, 256 threads
            int f   = i * 256 + t;
            int row = f >> 3;           // 8 float4 per 32-float chunk row
            int kq  = f & 7;
            float4 va = *(const float4*)(aBase + (size_t)row * D + kc + kq * 4);
            *(float4*)&lA[row * LSTR + kq * 4] = va;
            float4 vb = *(const float4*)(bBase + (size_t)row * D + kc + kq * 4);
            *(float4*)&lB[row * LSTR + kq * 4] = vb;
        }
        __syncthreads();

        #pragma unroll
        for (int k0 = 0; k0 < KC; k0 += 4) {
            const int kk = k0 + kl;
            v2f a[2], b[4];
            #pragma unroll
            for (int mt = 0; mt < 2; ++mt) {
                int r = waveM * 32 + mt * 16 + (lane & 15);
                a[mt] = *(const v2f*)&lA[r * LSTR + kk];
            }
            #pragma unroll
            for (int nt = 0; nt < 4; ++nt) {
                int c = waveN * 64 + nt * 16 + (lane & 15);
                b[nt] = *(const v2f*)&lB[c * LSTR + kk];
            }
            #pragma unroll
            for (int mt = 0; mt < 2; ++mt)
                #pragma unroll
                for (int nt = 0; nt < 4; ++nt)
                    acc[mt][nt] = __builtin_amdgcn_wmma_f32_16x16x4_f32(
                        false, a[mt], false, b[nt], (short)0, acc[mt][nt],
                        false, false);
        }
    }

    // epilogue: l2 = max(sq_i + sq_j - 2*g, 0); sum of 5 exp2(c_b * l2)
    float lsum = 0.f;
    #pragma unroll
    for (int mt = 0; mt < 2; ++mt) {
        #pragma unroll
        for (int nt = 0; nt < 4; ++nt) {
            const int colL = waveN * 64 + nt * 16 + (lane & 15);
            const float sqj = sqB[colL];
            #pragma unroll
            for (int e = 0; e < 8; ++e) {
                const int rowL = waveM * 32 + mt * 16 + e + ((lane & 16) ? 8 : 0);
                const float g  = acc[mt][nt][e];
                float l2 = sqA[rowL] + sqj - 2.0f * g;
                l2 = fmaxf(l2, 0.f);
                lsum += __builtin_amdgcn_exp2f(c0 * l2)
                      + __builtin_amdgcn_exp2f(c1 * l2)
                      + __builtin_amdgcn_exp2f(c2 * l2)
                      + __builtin_amdgcn_exp2f(c3 * l2)
                      + __builtin_amdgcn_exp2f(c4 * l2);
            }
        }
    }

    __syncthreads();
    red[t] = lsum;
    __syncthreads();
    for (int st = 128; st > 0; st >>= 1) {
        if (t < st) red[t] += red[t + st];
        __syncthreads();
    }
    if (t == 0) {
        const int regA = ((iBase >= NSRC) ? 2 : 0) | ((jBase >= NSRC) ? 1 : 0);
        atomicAdd(&ws[WS_SUMS + regA], red[0]);
        if (blockIdx.x != blockIdx.y) {   // mirror tile (j,i) has identical sum
            const int regB = ((jBase >= NSRC) ? 2 : 0) | ((iBase >= NSRC) ? 1 : 0);
            atomicAdd(&ws[WS_SUMS + regB], red[0]);
        }
    }
}

// ---------------- pass 4: combine region means ----------------
__global__ void finalize_kernel(const float* __restrict__ ws, float* __restrict__ out) {
    if (threadIdx.x == 0 && blockIdx.x == 0) {
        const double ns2 = (double)NSRC * (double)NSRC;
        const double nt2 = (double)NTGT * (double)NTGT;
        const double nst = (double)NSRC * (double)NTGT;
        double v = (double)ws[WS_SUMS + 0] / ns2 + (double)ws[WS_SUMS + 3] / nt2
                 - (double)ws[WS_SUMS + 1] / nst - (double)ws[WS_SUMS + 2] / nst;
        out[0] = (float)v;
    }
}

extern "C" void kernel_launch(void* const* d_in, const int* in_sizes, int n_in,
                              void* d_out, int out_size, void* d_ws, size_t ws_size,
                              hipStream_t stream) {
    (void)in_sizes; (void)n_in; (void)out_size; (void)ws_size;
    const float* src = (const float*)d_in[0];
    const float* tgt = (const float*)d_in[1];
    float* ws  = (float*)d_ws;
    float* out = (float*)d_out;

    sq_kernel<<<NTOT / 8, 256, 0, stream>>>(src, tgt, ws);
    bw_kernel<<<1, 256, 0, stream>>>(src, tgt, ws);
    dim3 grid(NTOT / BN, NTOT / BM);
    mmd_kernel<<<grid, 256, 0, stream>>>(src, tgt, ws);
    finalize_kernel<<<1, 64, 0, stream>>>(ws, out);
}